// HybridMemory_25658134626967
// MI455X (gfx1250) — compile-verified
//
#include <hip/hip_runtime.h>
#include <math.h>

#define NUM_MEMORY   100000
#define NUM_FEATURES 256
#define NUM_CLASSES  5000
#define BATCH        256
#define TEMP         0.05f
#define EPSV         1e-6f
#define SORT_CAP     1024     // per-class LDS sort capacity (E[count]=20; Poisson tail << cap)

typedef __attribute__((ext_vector_type(2))) float v2f;
typedef __attribute__((ext_vector_type(8))) float v8f;

// ---------------------------------------------------------------------------
// 0) zero the class histogram
// ---------------------------------------------------------------------------
__global__ void hm_zero_kernel(int* __restrict__ nums) {
    int i = blockIdx.x * blockDim.x + threadIdx.x;
    if (i < NUM_CLASSES) nums[i] = 0;
}

// ---------------------------------------------------------------------------
// 1) L2-normalize each batch row: x = inputs / max(||inputs||, 1e-12)
// ---------------------------------------------------------------------------
__global__ void hm_normalize_kernel(const float* __restrict__ in, float* __restrict__ x) {
    __shared__ float red[NUM_FEATURES];
    const int b = blockIdx.x, t = threadIdx.x;
    float v = in[b * NUM_FEATURES + t];
    red[t] = v * v;
    __syncthreads();
    for (int s = NUM_FEATURES / 2; s > 0; s >>= 1) {
        if (t < s) red[t] += red[t + s];
        __syncthreads();
    }
    float nrm = fmaxf(sqrtf(red[0]), 1e-12f);
    x[b * NUM_FEATURES + t] = v / nrm;
}

// ---------------------------------------------------------------------------
// 2a) label histogram: nums[labels[m]]++  (100K int atomics)
// ---------------------------------------------------------------------------
__global__ void hm_count_kernel(const int* __restrict__ labels, int* __restrict__ nums) {
    int m = blockIdx.x * blockDim.x + threadIdx.x;
    if (m < NUM_MEMORY) atomicAdd(&nums[labels[m]], 1);
}

// ---------------------------------------------------------------------------
// 2b) single-block exclusive scan of nums[5000] -> offsets, cursor
// ---------------------------------------------------------------------------
__global__ void hm_scan_kernel(const int* __restrict__ nums,
                               int* __restrict__ offsets, int* __restrict__ cursor) {
    __shared__ int part[256];
    const int t = threadIdx.x;
    const int per = (NUM_CLASSES + 255) / 256;   // 20
    const int base = t * per;
    int s = 0;
    for (int i = 0; i < per; ++i) {
        int c = base + i;
        if (c < NUM_CLASSES) s += nums[c];
    }
    part[t] = s;
    __syncthreads();
    // Hillis-Steele inclusive scan over per-thread partials
    for (int off = 1; off < 256; off <<= 1) {
        int v = (t >= off) ? part[t - off] : 0;
        __syncthreads();
        part[t] += v;
        __syncthreads();
    }
    int acc = part[t] - s;                        // exclusive prefix of this thread's span
    for (int i = 0; i < per; ++i) {
        int c = base + i;
        if (c < NUM_CLASSES) {
            offsets[c] = acc;
            cursor[c]  = acc;
            acc += nums[c];
        }
    }
}

// ---------------------------------------------------------------------------
// 2c) scatter row ids grouped by class (100K int atomics)
// ---------------------------------------------------------------------------
__global__ void hm_scatter_kernel(const int* __restrict__ labels,
                                  int* __restrict__ cursor, int* __restrict__ order) {
    int m = blockIdx.x * blockDim.x + threadIdx.x;
    if (m < NUM_MEMORY) {
        int pos = atomicAdd(&cursor[labels[m]], 1);
        order[pos] = m;
    }
}

// ---------------------------------------------------------------------------
// 2d) atomic-free per-class reduction: G[c,:] = sum of member feature rows.
//     One block (64 threads = 2 waves) per class; each thread owns one float4
//     column slice and register-accumulates over ~20 rows (1KB contiguous
//     coalesced loads). This is the single, unavoidable 102.4 MB pass over
//     the memory bank -> pure HBM-bandwidth bound (~4.4 us at 23.3 TB/s).
//     Indices are rank-sorted in LDS first so FP sum order is deterministic.
// ---------------------------------------------------------------------------
__global__ void hm_class_reduce_kernel(const float* __restrict__ feats,
                                       const int* __restrict__ nums,
                                       const int* __restrict__ offsets,
                                       const int* __restrict__ order,
                                       float* __restrict__ G) {
    __shared__ int sidx[SORT_CAP];
    __shared__ int ssorted[SORT_CAP];
    const int c = blockIdx.x, t = threadIdx.x;          // t in 0..63
    const int cnt   = nums[c];
    const int start = offsets[c];
    const int cs    = (cnt < SORT_CAP) ? cnt : SORT_CAP;

    for (int e = t; e < cs; e += 64) sidx[e] = order[start + e];
    __syncthreads();
    for (int e = t; e < cs; e += 64) {                  // rank sort (indices unique)
        int v = sidx[e], rank = 0;
        for (int j = 0; j < cs; ++j) rank += (sidx[j] < v);
        ssorted[rank] = v;
    }
    __syncthreads();

    float4 acc = make_float4(0.f, 0.f, 0.f, 0.f);
    for (int i = 0; i < cnt; ++i) {
        int m = (i < cs) ? ssorted[i] : order[start + i];
        float4 v = ((const float4*)(feats + (size_t)m * NUM_FEATURES))[t];
        acc.x += v.x; acc.y += v.y; acc.z += v.z; acc.w += v.w;
    }
    ((float4*)(G + (size_t)c * NUM_FEATURES))[t] = acc;  // empty class -> zeros
}

// ---------------------------------------------------------------------------
// 3) sim[b,c] = dot(x[b,:], G[c,:]) / (TEMP * max(nums[c],1))
//    M=256, N=5000, K=256 via V_WMMA_F32_16X16X4_F32; one wave per 16x16 tile.
// ---------------------------------------------------------------------------
__global__ void hm_gemm_wmma_kernel(const float* __restrict__ x,
                                    const float* __restrict__ G,
                                    const int* __restrict__ nums,
                                    float* __restrict__ sim) {
    const int NT_M = BATCH / 16;                 // 16
    const int NT_N = (NUM_CLASSES + 15) / 16;    // 313
    const int wave = blockIdx.x * (blockDim.x >> 5) + (threadIdx.x >> 5);
    const int lane = threadIdx.x & 31;
    const int tile_m = wave % NT_M;
    const int tile_n = wave / NT_M;
    if (tile_n >= NT_N) return;                  // wave-uniform (EXEC stays all-1)

    const int m0 = tile_m * 16;
    const int c0 = tile_n * 16;
    const int r     = lane & 15;
    const int khalf = (lane >> 4) << 1;          // lanes 0-15: K+0/1, lanes 16-31: K+2/3

    int colc = c0 + r;                           // clamp loads of partial last N-tile
    if (colc > NUM_CLASSES - 1) colc = NUM_CLASSES - 1;

    const float* arow = x + (size_t)(m0 + r) * NUM_FEATURES + khalf;
    const float* brow = G + (size_t)colc    * NUM_FEATURES + khalf;

    v8f acc = {};
    for (int k0 = 0; k0 < NUM_FEATURES; k0 += 4) {
        v2f a = *(const v2f*)(arow + k0);
        v2f b = *(const v2f*)(brow + k0);
        acc = __builtin_amdgcn_wmma_f32_16x16x4_f32(
            false, a, false, b, (short)0, acc, false, false);
    }

    const int col = c0 + r;
    if (col < NUM_CLASSES) {
        const int n = nums[col];
        const float scale = 1.0f / (TEMP * (float)(n > 0 ? n : 1));
        const int mbase = m0 + ((lane >> 4) << 3);   // D: lanes 16-31 hold M = i+8
#pragma unroll
        for (int i = 0; i < 8; ++i)
            sim[(size_t)(mbase + i) * NUM_CLASSES + col] = acc[i] * scale;
    }
}

// ---------------------------------------------------------------------------
// 4) per-row masked softmax denominator + target log-prob (reference: plain
//    exp, eps in denominator and inside log; sim is O(1) so no overflow)
// ---------------------------------------------------------------------------
__global__ void hm_loss_kernel(const float* __restrict__ sim,
                               const int* __restrict__ nums,
                               const int* __restrict__ indexes,
                               const int* __restrict__ labels,
                               float* __restrict__ rowloss) {
    __shared__ float red[256];
    const int b = blockIdx.x, t = threadIdx.x;
    float acc = 0.0f;
    for (int c = t; c < NUM_CLASSES; c += 256) {
        if (nums[c] > 0) acc += expf(sim[(size_t)b * NUM_CLASSES + c]);
    }
    red[t] = acc;
    __syncthreads();
    for (int s = 128; s > 0; s >>= 1) {
        if (t < s) red[t] += red[t + s];
        __syncthreads();
    }
    if (t == 0) {
        const int target = labels[indexes[b]];
        const float tv   = expf(sim[(size_t)b * NUM_CLASSES + target]);
        const float msim = tv / (red[0] + EPSV);
        rowloss[b] = -logf(msim + EPSV);
    }
}

// ---------------------------------------------------------------------------
// 5) mean over 256 per-row losses -> scalar
// ---------------------------------------------------------------------------
__global__ void hm_mean_kernel(const float* __restrict__ rowloss, float* __restrict__ out) {
    __shared__ float red[256];
    const int t = threadIdx.x;
    red[t] = rowloss[t];
    __syncthreads();
    for (int s = 128; s > 0; s >>= 1) {
        if (t < s) red[t] += red[t + s];
        __syncthreads();
    }
    if (t == 0) out[0] = red[0] / (float)BATCH;
}

// ---------------------------------------------------------------------------
extern "C" void kernel_launch(void* const* d_in, const int* in_sizes, int n_in,
                              void* d_out, int out_size, void* d_ws, size_t ws_size,
                              hipStream_t stream) {
    const float* inputs  = (const float*)d_in[0];   // [256,256]
    const int*   indexes = (const int*)  d_in[1];   // [256]
    const float* feats   = (const float*)d_in[2];   // [100000,256]
    const int*   labels  = (const int*)  d_in[3];   // [100000]

    // workspace layout (16B aligned), ~11.0 MB total
    int*   nums    = (int*)d_ws;                                 // 5000  (pad 5120)
    int*   offsets = nums + 5120;                                // 5000  (pad 5120)
    int*   cursor  = offsets + 5120;                             // 5000  (pad 5120)
    int*   order   = cursor + 5120;                              // 100000 (pad 100096)
    float* x       = (float*)(order + 100096);                   // 256*256
    float* G       = x + BATCH * NUM_FEATURES;                   // 5000*256
    float* sim     = G + NUM_CLASSES * NUM_FEATURES;             // 256*5000
    float* rowloss = sim + (size_t)BATCH * NUM_CLASSES;          // 256

    hm_zero_kernel<<<(NUM_CLASSES + 255) / 256, 256, 0, stream>>>(nums);
    hm_normalize_kernel<<<BATCH, NUM_FEATURES, 0, stream>>>(inputs, x);
    hm_count_kernel<<<(NUM_MEMORY + 255) / 256, 256, 0, stream>>>(labels, nums);
    hm_scan_kernel<<<1, 256, 0, stream>>>(nums, offsets, cursor);
    hm_scatter_kernel<<<(NUM_MEMORY + 255) / 256, 256, 0, stream>>>(labels, cursor, order);
    hm_class_reduce_kernel<<<NUM_CLASSES, 64, 0, stream>>>(feats, nums, offsets, order, G);
    hm_gemm_wmma_kernel<<<626, 256, 0, stream>>>(x, G, nums, sim);   // 5008 waves = 16x313 tiles
    hm_loss_kernel<<<BATCH, 256, 0, stream>>>(sim, nums, indexes, labels, rowloss);
    hm_mean_kernel<<<1, 256, 0, stream>>>(rowloss, (float*)d_out);
}